// NeuralHashVoxel_86938728006139
// MI455X (gfx1250) — compile-verified
//
#include <hip/hip_runtime.h>

typedef float v4f __attribute__((ext_vector_type(4)));

namespace {
constexpr int      LVL    = 6;
constexpr int      NF     = 8;          // features per entry
constexpr long long TBL   = 524288;     // entries per level
constexpr unsigned  BMASK = 4194304u - 1u; // hash buckets - 1 (2^22 - 1)
constexpr unsigned  P0 = 73856093u, P1 = 19349669u, P2 = 83492791u;
}

__device__ __forceinline__ unsigned hash3(unsigned cx, unsigned cy, unsigned cz) {
    // corners are non-negative and 2^22 | 2^32, so wrapping u32 arithmetic
    // masked by (B-1) matches the reference's int64 ((c*P).sum() mod B).
    return (cx * P0 + cy * P1 + cz * P2) & BMASK;
}

__global__ __launch_bounds__(256)
void NeuralHashVoxel_kernel(const float* __restrict__ q,      // (N,3) f32
                            const float* __restrict__ feat,   // (L,T,F) f32
                            const int*   __restrict__ fidx,   // (L,B) int64 viewed as i32 pairs
                            float* __restrict__ out,          // (N,F) f32
                            int n)
{
    const int tid = blockIdx.x * blockDim.x + threadIdx.x;
    if (tid >= n) return;

    // Streamed inputs: non-temporal so they don't evict the hot hash/feature
    // tables from the 192 MB L2.
    const float qx = __builtin_nontemporal_load(q + 3ull * (unsigned)tid + 0);
    const float qy = __builtin_nontemporal_load(q + 3ull * (unsigned)tid + 1);
    const float qz = __builtin_nontemporal_load(q + 3ull * (unsigned)tid + 2);

    // ---- Pass 1: prefetch all 48 hash-table entries (gfx1250 global_prefetch)
    // so the dependent random gathers below overlap instead of serializing.
#pragma unroll
    for (int i = 0; i < LVL; ++i) {
        const float invres = 4.0f / (float)(1 << i);   // exact power of two
        const unsigned bx = (unsigned)(int)floorf(qx * invres);
        const unsigned by = (unsigned)(int)floorf(qy * invres);
        const unsigned bz = (unsigned)(int)floorf(qz * invres);
        const int* tab = fidx + (size_t)i * (2ull * (size_t)(BMASK + 1u));
#pragma unroll
        for (int k = 0; k < 8; ++k) {
            const unsigned key = hash3(bx + ((k >> 2) & 1), by + ((k >> 1) & 1), bz + (k & 1));
            __builtin_prefetch(tab + 2ull * key, 0, 3);
        }
    }

    float a0 = 0.f, a1 = 0.f, a2 = 0.f, a3 = 0.f;
    float a4 = 0.f, a5 = 0.f, a6 = 0.f, a7 = 0.f;

    // ---- Pass 2: gather + trilinear blend, gated per lane on validity.
#pragma unroll
    for (int i = 0; i < LVL; ++i) {
        const float invres = 4.0f / (float)(1 << i);
        const float sx = qx * invres, sy = qy * invres, sz = qz * invres;
        const float fx = floorf(sx), fy = floorf(sy), fz = floorf(sz);
        const float tx = sx - fx,    ty = sy - fy,    tz = sz - fz;
        const unsigned bx = (unsigned)(int)fx;
        const unsigned by = (unsigned)(int)fy;
        const unsigned bz = (unsigned)(int)fz;

        const int* tab = fidx + (size_t)i * (2ull * (size_t)(BMASK + 1u));
        int id[8];
#pragma unroll
        for (int k = 0; k < 8; ++k) {
            const unsigned key = hash3(bx + ((k >> 2) & 1), by + ((k >> 1) & 1), bz + (k & 1));
            id[k] = tab[2ull * key];   // low dword of the int64 entry: -1 or [0, T)
        }

        // Level contributes iff every corner is non-empty (entry >= 0).
        // Sign-bit OR is the cheapest all-non-negative test; EXEC-masks the
        // 8x32B feature gathers away for the ~83% of invalid (point,level)s.
        const int sgn = id[0] | id[1] | id[2] | id[3] | id[4] | id[5] | id[6] | id[7];
        if (sgn >= 0) {
            const float* fl = feat + (size_t)i * ((size_t)TBL * NF);
            const float w1x = tx, w0x = 1.0f - tx;
            const float w1y = ty, w0y = 1.0f - ty;
            const float w1z = tz, w0z = 1.0f - tz;
#pragma unroll
            for (int k = 0; k < 8; ++k) {
                const float c = (((k & 4) ? w1x : w0x) * ((k & 2) ? w1y : w0y))
                                 * ((k & 1) ? w1z : w0z);
                const v4f* p = (const v4f*)(fl + (size_t)id[k] * NF); // 16B aligned
                const v4f lo = p[0];   // global_load_b128
                const v4f hi = p[1];   // global_load_b128
                a0 = fmaf(c, lo[0], a0); a1 = fmaf(c, lo[1], a1);
                a2 = fmaf(c, lo[2], a2); a3 = fmaf(c, lo[3], a3);
                a4 = fmaf(c, hi[0], a4); a5 = fmaf(c, hi[1], a5);
                a6 = fmaf(c, hi[2], a6); a7 = fmaf(c, hi[3], a7);
            }
        }
    }

    // Streamed output: NT b128 stores (32 MB written once, keep L2 for tables).
    v4f o0 = {a0, a1, a2, a3};
    v4f o1 = {a4, a5, a6, a7};
    v4f* dst = (v4f*)(out + (size_t)tid * NF);
    __builtin_nontemporal_store(o0, dst);
    __builtin_nontemporal_store(o1, dst + 1);
}

extern "C" void kernel_launch(void* const* d_in, const int* in_sizes, int n_in,
                              void* d_out, int out_size, void* d_ws, size_t ws_size,
                              hipStream_t stream) {
    (void)n_in; (void)d_ws; (void)ws_size; (void)out_size;
    const float* q    = (const float*)d_in[0];   // query_points (N,3) f32
    const float* feat = (const float*)d_in[1];   // features (L,T,F) f32
    const int*   fidx = (const int*)d_in[2];     // feature_indexs (L,B) int64 -> i32 pairs
    float*       out  = (float*)d_out;           // (N,F) f32

    const int n = in_sizes[0] / 3;               // number of query points
    const int threads = 256;                     // 8 wave32s per block
    const int blocks  = (n + threads - 1) / threads;
    NeuralHashVoxel_kernel<<<blocks, threads, 0, stream>>>(q, feat, fidx, out, n);
}